// Attention_48962627174903
// MI455X (gfx1250) — compile-verified
//
#include <hip/hip_runtime.h>
#include <math.h>
#include <stdint.h>

typedef __attribute__((ext_vector_type(2))) float v2f;
typedef __attribute__((ext_vector_type(8))) float v8f;

constexpr int B  = 16;
constexpr int C  = 64;
constexpr int H  = 128;
constexpr int W  = 128;
constexpr int CQ = 8;

// ---------------------------------------------------------------------------
// K1: per-pixel 1x1 conv projections (VALU; bandwidth-bound part).
// Writes q_t,k_t as [b][w][t][c] (c fastest -> 8B-contiguous WMMA fragments)
// and v as [b][w][h][c] (256B contiguous per thread).
// ---------------------------------------------------------------------------
__global__ __launch_bounds__(128) void k1_proj(
    const float* __restrict__ x,
    const float* __restrict__ wq, const float* __restrict__ bq,
    const float* __restrict__ wk, const float* __restrict__ bk,
    const float* __restrict__ wv, const float* __restrict__ bv,
    float* __restrict__ q_t, float* __restrict__ k_t, float* __restrict__ v_t)
{
    __shared__ float xs[C * W];          // 32 KB: x[b, :, h, :]
    __shared__ float wqs[CQ * C], wks[CQ * C];
    __shared__ float wvs[C * C];
    __shared__ float bqs[CQ], bks[CQ], bvs[C];

    const int h = blockIdx.x;
    const int b = blockIdx.y;
    const int tid = threadIdx.x;

    for (int i = tid; i < C * W; i += 128) {
        int c = i >> 7, w = i & 127;
        xs[i] = x[(((size_t)b * C + c) * H + h) * W + w];
    }
    for (int i = tid; i < CQ * C; i += 128) { wqs[i] = wq[i]; wks[i] = wk[i]; }
    for (int i = tid; i < C * C;  i += 128) { wvs[i] = wv[i]; }
    if (tid < CQ) { bqs[tid] = bq[tid]; bks[tid] = bk[tid]; }
    if (tid < C)  { bvs[tid] = bv[tid]; }
    __syncthreads();

    const int w = tid;                    // one pixel (h,w) per thread
    float xv[C];
    #pragma unroll
    for (int c = 0; c < C; ++c) xv[c] = xs[c * W + w];

    const size_t pix = ((size_t)(b * W + w)) * H + h;
    float* qo = q_t + pix * CQ;
    float* ko = k_t + pix * CQ;
    float* vo = v_t + pix * C;

    #pragma unroll
    for (int o = 0; o < CQ; ++o) {
        float aq = bqs[o], ak = bks[o];
        #pragma unroll
        for (int c = 0; c < C; ++c) {
            aq = fmaf(wqs[o * C + c], xv[c], aq);
            ak = fmaf(wks[o * C + c], xv[c], ak);
        }
        qo[o] = aq;
        ko[o] = ak;
    }
    for (int o = 0; o < C; ++o) {
        float av = bvs[o];
        #pragma unroll
        for (int c = 0; c < C; ++c) av = fmaf(wvs[o * C + c], xv[c], av);
        vo[o] = av;
    }
}

// ---------------------------------------------------------------------------
// K2: energy GEMM, one workgroup per (b,w).
// E_w[t,h] = sum_{c=0..7} Q_w[t,c] * K_w[h,c]  via V_WMMA_F32_16X16X4_F32.
// Wave = one 16-row t-tile, iterates 8 h-tiles x 2 K-steps.
// ---------------------------------------------------------------------------
__global__ __launch_bounds__(256) void k2_energy(
    const float* __restrict__ q_t, const float* __restrict__ k_t,
    float* __restrict__ E)
{
    const int w = blockIdx.x, b = blockIdx.y;
    const int lane = threadIdx.x & 31;
    const int wave = threadIdx.x >> 5;
    const int t0   = wave * 16;
    const int l15  = lane & 15;
    const int sh   = (lane < 16) ? 0 : 2;    // K-half per ISA A/B f32 layout

    const float* qp = q_t + ((size_t)(b * W + w)) * H * CQ;
    const float* kp = k_t + ((size_t)(b * W + w)) * H * CQ;
    float*       ep = E   + ((size_t)(b * W + w)) * H * H;

    v8f acc[8];
    #pragma unroll
    for (int i = 0; i < 8; ++i)
        #pragma unroll
        for (int r = 0; r < 8; ++r) acc[i][r] = 0.0f;

    #pragma unroll
    for (int c0 = 0; c0 < CQ; c0 += 4) {
        // A: 16x4, lane holds M=t0+l15, K = c0+sh, c0+sh+1 (8B contiguous)
        v2f a = *(const v2f*)(qp + (t0 + l15) * CQ + c0 + sh);
        #pragma unroll
        for (int ht = 0; ht < 8; ++ht) {
            v2f bf = *(const v2f*)(kp + (ht * 16 + l15) * CQ + c0 + sh);
            acc[ht] = __builtin_amdgcn_wmma_f32_16x16x4_f32(
                false, a, false, bf, (short)0, acc[ht], false, false);
        }
    }

    const int trow = t0 + ((lane >= 16) ? 8 : 0);
    #pragma unroll
    for (int ht = 0; ht < 8; ++ht)
        #pragma unroll
        for (int r = 0; r < 8; ++r)
            ep[(trow + r) * H + ht * 16 + l15] = acc[ht][r];
}

// ---------------------------------------------------------------------------
// K3: softmax over w (in place). Plane layout [b][w][t*128+h].
// Workgroup owns 128 consecutive (t,h) columns; thread owns one column,
// stages its 128 w-values via coalesced loads into LDS. No barriers needed.
// ---------------------------------------------------------------------------
__global__ __launch_bounds__(128) void k3_softmax(float* __restrict__ E)
{
    __shared__ float se[W * 128];        // 64 KB: [w][th_local]
    const int thb = blockIdx.x, b = blockIdx.y;
    const int tid = threadIdx.x;
    const int th0 = thb * 128;
    const size_t base = (size_t)b * W * H * H;

    for (int w = 0; w < W; ++w)
        se[w * 128 + tid] = E[base + (size_t)w * H * H + th0 + tid];

    float m = -INFINITY;
    for (int w = 0; w < W; ++w) m = fmaxf(m, se[w * 128 + tid]);
    float s = 0.0f;
    for (int w = 0; w < W; ++w) {
        float e = __expf(se[w * 128 + tid] - m);
        s += e;
        se[w * 128 + tid] = e;
    }
    const float inv = 1.0f / s;
    for (int w = 0; w < W; ++w)
        E[base + (size_t)w * H * H + th0 + tid] = se[w * 128 + tid] * inv;
}

// ---------------------------------------------------------------------------
// K4: out GEMM, one workgroup per (b,w).
// D_w[t,c] = sum_h attn_w[t,h] * V_w[h,c], K=128 via 32 chained K4 WMMAs.
// V plane (32 KB, contiguous) staged into LDS with CDNA5 async-LDS loads
// (global_load_async_to_lds_b128, tracked by ASYNCcnt), then consumed as
// B fragments straight from LDS. Epilogue staged through LDS ->
// fully coalesced out_tmp[b][w][t][c] stores.
// ---------------------------------------------------------------------------
__global__ __launch_bounds__(256) void k4_out(
    const float* __restrict__ attn, const float* __restrict__ v_t,
    float* __restrict__ out_tmp)
{
    __shared__ float Vs[H * C];          // 32 KB, plain [h][c] layout
    __shared__ float Os[8 * 16 * C];     // 32 KB, per-wave [16t][64c] staging

    const int w = blockIdx.x, b = blockIdx.y;
    const size_t plane = (size_t)(b * W + w);
    const float* vp = v_t  + plane * H * C;
    const float* ap = attn + plane * H * H;
    float*       op = out_tmp + plane * H * C;

    // --- async identity copy: 32 KB global -> LDS, 16B per lane per issue ---
    {
        const uint32_t lds_base = (uint32_t)(uintptr_t)(&Vs[0]); // LDS offset = low 32 bits of the pointer
        const uint64_t gbase    = (uint64_t)(uintptr_t)vp;
        #pragma unroll
        for (int it = 0; it < 8; ++it) {
            uint32_t off = ((uint32_t)threadIdx.x + it * 256u) * 16u;
            uint32_t la  = lds_base + off;
            uint64_t ga  = gbase + off;
            asm volatile("global_load_async_to_lds_b128 %0, %1, off"
                         :: "v"(la), "v"(ga) : "memory");
        }
        asm volatile("s_wait_asynccnt 0" ::: "memory");
    }
    __syncthreads();

    const int lane = threadIdx.x & 31;
    const int wave = threadIdx.x >> 5;
    const int t0   = wave * 16;
    const int l15  = lane & 15;
    const int sh   = (lane < 16) ? 0 : 2;

    // hint: this wave's attn row block will be streamed sequentially
    __builtin_prefetch(ap + (t0 + l15) * H, 0, 1);

    v8f acc[4];
    #pragma unroll
    for (int i = 0; i < 4; ++i)
        #pragma unroll
        for (int r = 0; r < 8; ++r) acc[i][r] = 0.0f;

    for (int h0 = 0; h0 < H; h0 += 4) {
        // A: attn[t0+l15][h0+sh .. +1], 8B contiguous global load (L0 reuse)
        v2f a = *(const v2f*)(ap + (t0 + l15) * H + h0 + sh);
        const int hh = h0 + sh;
        #pragma unroll
        for (int ct = 0; ct < 4; ++ct) {
            v2f bf;
            bf.x = Vs[(hh + 0) * C + ct * 16 + l15];  // B VGPR0: K=hh
            bf.y = Vs[(hh + 1) * C + ct * 16 + l15];  // B VGPR1: K=hh+1
            acc[ct] = __builtin_amdgcn_wmma_f32_16x16x4_f32(
                false, a, false, bf, (short)0, acc[ct], false, false);
        }
    }

    float* os = Os + wave * (16 * C);
    const int trow = ((lane >= 16) ? 8 : 0);
    #pragma unroll
    for (int ct = 0; ct < 4; ++ct)
        #pragma unroll
        for (int r = 0; r < 8; ++r)
            os[(trow + r) * C + ct * 16 + l15] = acc[ct][r];
    // same-wave DS RAW tracked by DScnt (compiler inserts waits)

    float4*       dst = (float4*)(op + t0 * C);
    const float4* src = (const float4*)os;
    for (int i = lane; i < (16 * C) / 4; i += 32)
        dst[i] = src[i];
}

// ---------------------------------------------------------------------------
// K5: out[b,c,t,w] = gamma * out_tmp[b][w][t][c] + x[b,c,t,w]
// 32x32 LDS-tiled (w,c)->(c,w) permute; both sides 128B-coalesced.
// ---------------------------------------------------------------------------
__global__ __launch_bounds__(256) void k5_final(
    const float* __restrict__ out_tmp, const float* __restrict__ x,
    const float* __restrict__ gamma, float* __restrict__ out)
{
    __shared__ float tile[32 * 33];
    const int t  = blockIdx.y;
    const int b  = blockIdx.z;
    const int c0 = (blockIdx.x >> 2) * 32;   // 2 c-tiles
    const int w0 = (blockIdx.x & 3) * 32;    // 4 w-tiles
    const float g = gamma[0];

    for (int i = threadIdx.x; i < 1024; i += 256) {
        int cl = i & 31, wl = i >> 5;        // c fastest: coalesced read
        tile[wl * 33 + cl] =
            out_tmp[(((size_t)(b * W + (w0 + wl))) * H + t) * C + c0 + cl];
    }
    __syncthreads();
    for (int i = threadIdx.x; i < 1024; i += 256) {
        int wl = i & 31, cl = i >> 5;        // w fastest: coalesced write
        size_t oi = (((size_t)(b * C + c0 + cl)) * H + t) * W + w0 + wl;
        out[oi] = g * tile[wl * 33 + cl] + x[oi];
    }
}

// ---------------------------------------------------------------------------
extern "C" void kernel_launch(void* const* d_in, const int* in_sizes, int n_in,
                              void* d_out, int out_size, void* d_ws, size_t ws_size,
                              hipStream_t stream)
{
    const float* x     = (const float*)d_in[0];
    const float* wq    = (const float*)d_in[1];
    const float* bq    = (const float*)d_in[2];
    const float* wk    = (const float*)d_in[3];
    const float* bk    = (const float*)d_in[4];
    const float* wv    = (const float*)d_in[5];
    const float* bv    = (const float*)d_in[6];
    const float* gamma = (const float*)d_in[7];
    float* out = (float*)d_out;

    float* ws      = (float*)d_ws;
    float* q_t     = ws;                                   //  8 MB
    float* k_t     = q_t  + (size_t)B * W * H * CQ;        //  8 MB
    float* v_t     = k_t  + (size_t)B * W * H * CQ;        // 64 MB
    float* E       = v_t  + (size_t)B * W * H * C;         // 134 MB (E, then attn in place)
    float* out_tmp = E    + (size_t)B * W * H * H;         // 64 MB

    k1_proj   <<<dim3(H, B),           128, 0, stream>>>(x, wq, bq, wk, bk, wv, bv, q_t, k_t, v_t);
    k2_energy <<<dim3(W, B),           256, 0, stream>>>(q_t, k_t, E);
    k3_softmax<<<dim3(H * H / 128, B), 128, 0, stream>>>(E);
    k4_out    <<<dim3(W, B),           256, 0, stream>>>(E, v_t, out_tmp);
    k5_final  <<<dim3(8, H, B),        256, 0, stream>>>(out_tmp, x, gamma, out);
}